// TripletModel_4234837754127
// MI455X (gfx1250) — compile-verified
//
#include <hip/hip_runtime.h>

typedef __attribute__((ext_vector_type(2))) float v2f;
typedef __attribute__((ext_vector_type(8))) float v8f;

#define TD 384              // feature dimension D
#define WAVES_PER_BLOCK 8
#define ROWS_PER_WAVE 16
#define ROWS_PER_BLOCK (WAVES_PER_BLOCK * ROWS_PER_WAVE)  // 128
#define TALPHA 1.0f

// ---------------------------------------------------------------------------
// Main kernel: each wave owns 16 rows. Lane L accumulates the partial
// d = sum_c w[c] * ((a-p)^2 - (a-n)^2) for row (L%16), with lanes L and L+16
// covering interleaved float4 column chunks of the same row. A single f32
// WMMA (16x16x4) against a ones-B matrix then sums the lane pairs, producing
// all 16 exact f32 row sums in one matrix op.
// ---------------------------------------------------------------------------
__global__ __launch_bounds__(256) void triplet_rows_kernel(
    const float* __restrict__ anc,
    const float* __restrict__ pos,
    const float* __restrict__ neg,
    const float* __restrict__ W,
    float* __restrict__ block_sums)
{
    __shared__ float wexp[TD];
    __shared__ float wave_sum[WAVES_PER_BLOCK];

    // Stage exp(W) into LDS once per block (1.5 KB).
    for (int i = threadIdx.x; i < TD; i += blockDim.x)
        wexp[i] = __expf(W[i]);
    __syncthreads();

    const int tid  = threadIdx.x;
    const int wid  = tid >> 5;        // wave id in block (0..7)
    const int lane = tid & 31;        // lane in wave
    const int m    = lane & 15;       // row slot within the 16-row tile
    const int h    = lane >> 4;       // column phase: 0 or 1

    const long row = (long)blockIdx.x * ROWS_PER_BLOCK +
                     (long)wid * ROWS_PER_WAVE + m;
    const float* __restrict__ ar = anc + row * TD;
    const float* __restrict__ pr = pos + row * TD;
    const float* __restrict__ nr = neg + row * TD;

    float acc = 0.0f;
    // Lane pair (L, L+16) covers columns {c..c+3} and {c+4..c+7} per step.
    #pragma unroll 4
    for (int c = h * 4; c < TD; c += 8) {
        const float4 av = *(const float4*)(ar + c);
        const float4 pv = *(const float4*)(pr + c);
        const float4 nv = *(const float4*)(nr + c);
        const float4 wv = *(const float4*)(&wexp[c]);

        float dp, dn;
        dp = av.x - pv.x; dn = av.x - nv.x; acc += wv.x * (dp * dp - dn * dn);
        dp = av.y - pv.y; dn = av.y - nv.y; acc += wv.y * (dp * dp - dn * dn);
        dp = av.z - pv.z; dn = av.z - nv.z; acc += wv.z * (dp * dp - dn * dn);
        dp = av.w - pv.w; dn = av.w - nv.w; acc += wv.w * (dp * dp - dn * dn);
    }

    // Cross-lane pair reduction via f32 WMMA:
    //   A (16x4): lane L, VGPR0 -> (M=L%16, K=2*(L/16)); VGPR1 zeroed (K=1,3).
    //   B (4x16): all ones  =>  D[m][n] = acc(lane m) + acc(lane m+16).
    v2f A; A.x = acc;  A.y = 0.0f;
    v2f B; B.x = 1.0f; B.y = 1.0f;
    v8f C = {};
    v8f Dm = __builtin_amdgcn_wmma_f32_16x16x4_f32(
        /*neg_a=*/false, A, /*neg_b=*/false, B,
        /*c_mod=*/(short)0, C, /*reuse_a=*/false, /*reuse_b=*/false);

    // D layout: lane L, VGPR v -> (M = v + 8*(L/16), N = L%16); row sums are
    // replicated across all 16 columns. Fold relu(d + alpha) over this lane's
    // 8 rows, then add the other half-wave's 8 rows via a lane-16 xor shuffle.
    float local = 0.0f;
    #pragma unroll
    for (int v = 0; v < 8; ++v)
        local += fmaxf(Dm[v] + TALPHA, 0.0f);
    local += __shfl_xor(local, 16, 32);   // all lanes now hold 16-row total

    if (lane == 0) wave_sum[wid] = local;
    __syncthreads();

    if (tid == 0) {
        float s = 0.0f;
        #pragma unroll
        for (int i = 0; i < WAVES_PER_BLOCK; ++i) s += wave_sum[i];
        block_sums[blockIdx.x] = s;
    }
}

// ---------------------------------------------------------------------------
// Finalize: reduce per-block partials, write the mean.
// ---------------------------------------------------------------------------
__global__ __launch_bounds__(256) void triplet_finalize_kernel(
    const float* __restrict__ block_sums, int nblocks,
    float* __restrict__ out, float inv_n)
{
    __shared__ float sm[256];
    float s = 0.0f;
    for (int i = threadIdx.x; i < nblocks; i += 256)
        s += block_sums[i];
    sm[threadIdx.x] = s;
    __syncthreads();
    #pragma unroll
    for (int off = 128; off > 0; off >>= 1) {
        if (threadIdx.x < off) sm[threadIdx.x] += sm[threadIdx.x + off];
        __syncthreads();
    }
    if (threadIdx.x == 0) out[0] = sm[0] * inv_n;
}

extern "C" void kernel_launch(void* const* d_in, const int* in_sizes, int n_in,
                              void* d_out, int out_size, void* d_ws, size_t ws_size,
                              hipStream_t stream)
{
    const float* anc = (const float*)d_in[0];
    const float* pos = (const float*)d_in[1];
    const float* neg = (const float*)d_in[2];
    const float* W   = (const float*)d_in[3];
    float* out = (float*)d_out;

    const int N = in_sizes[0] / TD;               // 131072
    const int nblocks = N / ROWS_PER_BLOCK;       // 1024

    float* block_sums = (float*)d_ws;             // nblocks * 4 bytes

    triplet_rows_kernel<<<nblocks, 256, 0, stream>>>(anc, pos, neg, W, block_sums);
    triplet_finalize_kernel<<<1, 256, 0, stream>>>(block_sums, nblocks, out,
                                                   1.0f / (float)N);
}